// abggcn_53575422050318
// MI455X (gfx1250) — compile-verified
//
#include <hip/hip_runtime.h>

#define Bsz  16
#define Nn   1200
#define FINd 68
#define Dd   140
#define DPad 144
#define DFd  128
#define NGl  4

typedef float v2f __attribute__((ext_vector_type(2)));
typedef float v8f __attribute__((ext_vector_type(8)));

// ---------------------------------------------------------------------------
// Generic NT GEMM via V_WMMA_F32_16X16X4_F32:
//   C[M x Nc] = A[M x K] * B[Nc x K]^T (+ bias[col]) (+relu)
// Row-major, grid = (M/16, Nc/16, batch), block = 32 (one wave).
// f32 16x16x4 fragment layout (ISA 7.12.2): lane l holds A[row=l&15][k0+2*(l>>4)+{0,1}]
// and B-frag is loaded identically from the (transposed-operand) row-major B.
// ---------------------------------------------------------------------------
__global__ __launch_bounds__(32)
void wmma_nt_gemm(const float* __restrict__ A, const float* __restrict__ Bm,
                  float* __restrict__ C, const float* __restrict__ bias,
                  int K, int lda, int ldb, int ldc,
                  long long sA, long long sB, long long sC, int relu)
{
    const int lane = threadIdx.x;
    const int r    = lane & 15;
    const int hi   = lane >> 4;
    const float* Ab = A  + (long long)blockIdx.z * sA + (long long)(blockIdx.x * 16 + r) * lda + 2 * hi;
    const float* Bb = Bm + (long long)blockIdx.z * sB + (long long)(blockIdx.y * 16 + r) * ldb + 2 * hi;
    v8f acc = {0.f, 0.f, 0.f, 0.f, 0.f, 0.f, 0.f, 0.f};
    for (int k0 = 0; k0 < K; k0 += 4) {
        v2f a = *(const v2f*)(Ab + k0);
        v2f b = *(const v2f*)(Bb + k0);
        acc = __builtin_amdgcn_wmma_f32_16x16x4_f32(false, a, false, b, (short)0, acc, false, false);
    }
    const int col = blockIdx.y * 16 + r;
    const float bv = bias ? bias[col] : 0.0f;
    float* Cp = C + (long long)blockIdx.z * sC + (long long)(blockIdx.x * 16 + 8 * hi) * ldc + col;
#pragma unroll
    for (int v = 0; v < 8; ++v) {
        float val = acc[v] + bv;
        if (relu) val = fmaxf(val, 0.f);
        Cp[(long long)v * ldc] = val;
    }
}

// ---------------------------------------------------------------------------
// Fused symmetric attention-logit GEMM:
//   E[i,j] = hA[i]·h[j] + h[i]·hA[j]   ( == (hA hᵀ) + (hA hᵀ)ᵀ )
// Two WMMA accumulation chains into one accumulator; eliminates the separate
// e+eᵀ pass over the 92MB tensor entirely.  grid = (N/16, N/16, batch).
// ---------------------------------------------------------------------------
__global__ __launch_bounds__(32)
void wmma_nt_sym_gemm(const float* __restrict__ HAm, const float* __restrict__ Hm,
                      float* __restrict__ C, int K, int ld, int ldc,
                      long long sIn, long long sC)
{
    const int lane = threadIdx.x;
    const int r    = lane & 15;
    const int hi   = lane >> 4;
    const long long offI = (long long)(blockIdx.x * 16 + r) * ld + 2 * hi; // rows i
    const long long offJ = (long long)(blockIdx.y * 16 + r) * ld + 2 * hi; // rows j
    const float* haI = HAm + (long long)blockIdx.z * sIn + offI;
    const float* hJ  = Hm  + (long long)blockIdx.z * sIn + offJ;
    const float* hI  = Hm  + (long long)blockIdx.z * sIn + offI;
    const float* haJ = HAm + (long long)blockIdx.z * sIn + offJ;
    v8f acc = {0.f, 0.f, 0.f, 0.f, 0.f, 0.f, 0.f, 0.f};
    for (int k0 = 0; k0 < K; k0 += 4) {
        v2f a1 = *(const v2f*)(haI + k0);
        v2f b1 = *(const v2f*)(hJ + k0);
        acc = __builtin_amdgcn_wmma_f32_16x16x4_f32(false, a1, false, b1, (short)0, acc, false, false);
        v2f a2 = *(const v2f*)(hI + k0);
        v2f b2 = *(const v2f*)(haJ + k0);
        acc = __builtin_amdgcn_wmma_f32_16x16x4_f32(false, a2, false, b2, (short)0, acc, false, false);
    }
    const int col = blockIdx.y * 16 + r;
    float* Cp = C + (long long)blockIdx.z * sC + (long long)(blockIdx.x * 16 + 8 * hi) * ldc + col;
#pragma unroll
    for (int v = 0; v < 8; ++v)
        Cp[(long long)v * ldc] = acc[v];
}

// ---------------------------------------------------------------------------
// Generic NN GEMM via WMMA: C[M x Nc] = A[M x K] * B[K x Nc] (+relu)
// B fragment: lane l holds B[k0+2*(l>>4)+{0,1}][col=tile_n*16 + (l&15)]
// ---------------------------------------------------------------------------
__global__ __launch_bounds__(32)
void wmma_nn_gemm(const float* __restrict__ A, const float* __restrict__ Bm,
                  float* __restrict__ C,
                  int K, int lda, int ldb, int ldc,
                  long long sA, long long sB, long long sC, int relu)
{
    const int lane = threadIdx.x;
    const int r    = lane & 15;
    const int hi   = lane >> 4;
    const float* Ab = A  + (long long)blockIdx.z * sA + (long long)(blockIdx.x * 16 + r) * lda + 2 * hi;
    const float* Bb = Bm + (long long)blockIdx.z * sB + (long long)(blockIdx.y * 16 + r);
    v8f acc = {0.f, 0.f, 0.f, 0.f, 0.f, 0.f, 0.f, 0.f};
    for (int k0 = 0; k0 < K; k0 += 4) {
        v2f a = *(const v2f*)(Ab + k0);
        const float* bp = Bb + (long long)(k0 + 2 * hi) * ldb;
        v2f b;
        b.x = bp[0];
        b.y = bp[ldb];
        acc = __builtin_amdgcn_wmma_f32_16x16x4_f32(false, a, false, b, (short)0, acc, false, false);
    }
    const int col = blockIdx.y * 16 + r;
    float* Cp = C + (long long)blockIdx.z * sC + (long long)(blockIdx.x * 16 + 8 * hi) * ldc + col;
#pragma unroll
    for (int v = 0; v < 8; ++v) {
        float val = acc[v];
        if (relu) val = fmaxf(val, 0.f);
        Cp[(long long)v * ldc] = val;
    }
}

__device__ inline void sm_merge(float& m, float& s, float m2, float s2)
{
    if (s2 == 0.f) return;
    if (s == 0.f) { m = m2; s = s2; return; }
    if (m2 > m) { s = s * __expf(m - m2) + s2; m = m2; }
    else        { s = s + s2 * __expf(m2 - m); }
}

// --------------------------------------------------------------------------
// Masked softmax over axis=1 (columns of e), then multiply by edge weight.
// mode 0: mask = adj1>0, weight = adj1.  mode 1: unmasked, weight = RBF(d)+adj1.
// Block: 32 columns x 8 row-threads.  Result written in place into E.
// --------------------------------------------------------------------------
__global__ __launch_bounds__(256)
void col_softmax(float* __restrict__ E, const float* __restrict__ adj1,
                 const float* __restrict__ adj2d, const float* __restrict__ muP,
                 const float* __restrict__ devP, int mode)
{
    __shared__ float smM[8][32];
    __shared__ float smS[8][32];
    const int tx = threadIdx.x, ty = threadIdx.y;
    const int j = blockIdx.x * 32 + tx;
    const int b = blockIdx.y;
    const long long base = (long long)b * Nn * Nn;
    float m = -3.0e38f, s = 0.f;
    if (j < Nn) {
        for (int i = ty; i < Nn; i += 8) {
            long long off = base + (long long)i * Nn + j;
            float ev = E[off];
            bool ok = (mode == 0) ? (adj1[off] > 0.f) : true;
            if (ok) {
                if (ev > m) { s = s * __expf(m - ev) + 1.f; m = ev; }
                else        { s += __expf(ev - m); }
            }
        }
    }
    smM[ty][tx] = m; smS[ty][tx] = s;
    __syncthreads();
    if (ty == 0) {
        for (int t = 1; t < 8; ++t) sm_merge(m, s, smM[t][tx], smS[t][tx]);
        smM[0][tx] = m; smS[0][tx] = s;
    }
    __syncthreads();
    m = smM[0][tx];
    const float invS = 1.f / smS[0][tx];
    if (j < Nn) {
        const float mu = muP[0], dv = devP[0];
        for (int i = ty; i < Nn; i += 8) {
            long long off = base + (long long)i * Nn + j;
            float ev = E[off];
            float outv;
            if (mode == 0) {
                float a = adj1[off];
                outv = (a > 0.f) ? __expf(ev - m) * invS * a : 0.f;
            } else {
                float dd = adj2d[off] - mu;
                float w  = __expf(-dd * dd / dv) + adj1[off];
                outv = __expf(ev - m) * invS * w;
            }
            E[off] = outv;
        }
    }
}

// --------------------------------------------------------------------------
// coeff = sigmoid([x, h'] @ gw^T + gb);  out = coeff*x + (1-coeff)*h'
// one wave per (b,n) row
// --------------------------------------------------------------------------
__global__ __launch_bounds__(32)
void gate_combine(const float* __restrict__ X, const float* __restrict__ HP,
                  float* __restrict__ Out, const float* __restrict__ gw,
                  const float* __restrict__ gb)
{
    const int row  = blockIdx.x;
    const int lane = threadIdx.x;
    const float* x  = X  + (long long)row * DPad;
    const float* hp = HP + (long long)row * DPad;
    float acc = 0.f;
    for (int f = lane; f < Dd; f += 32) acc += x[f] * gw[f] + hp[f] * gw[Dd + f];
    for (int off = 16; off > 0; off >>= 1) acc += __shfl_xor(acc, off, 32);
    const float c = 1.f / (1.f + __expf(-(acc + gb[0])));
    float* o = Out + (long long)row * DPad;
    for (int f = lane; f < DPad; f += 32) o[f] = c * x[f] + (1.f - c) * hp[f];
}

__global__ void sub_kernel(const float* __restrict__ A, const float* __restrict__ Bm,
                           float* __restrict__ C, long long n)
{
    long long i = (long long)blockIdx.x * blockDim.x + threadIdx.x;
    if (i < n) C[i] = A[i] - Bm[i];
}

// --------------------------------------------------------------------------
// Pooling + BN/sigmoid:
//   pool_att[b,d] = mean_n( att_c[b,n,d] * valid[b,n] )
//   att = sigmoid(BN_n(att_c));  pool_xp[b,d] = sum_n x*(1+att)*valid
// --------------------------------------------------------------------------
__global__ __launch_bounds__(DPad)
void pool_kernel(const float* __restrict__ AC, const float* __restrict__ X,
                 const float* __restrict__ valid,
                 const float* __restrict__ bn_g, const float* __restrict__ bn_b,
                 const float* __restrict__ bn_m, const float* __restrict__ bn_v,
                 float* __restrict__ pool_att, float* __restrict__ pool_xp)
{
    const int b = blockIdx.x;
    const int d = threadIdx.x;
    float s_att = 0.f, s_xp = 0.f;
    for (int n = 0; n < Nn; ++n) {
        float vm = valid[b * Nn + n];
        long long off = ((long long)b * Nn + n) * DPad + d;
        float ac = AC[off];
        float xv = X[off];
        float inv = rsqrtf(bn_v[n] + 1e-5f);
        float att = 1.f / (1.f + __expf(-((ac - bn_m[n]) * inv * bn_g[n] + bn_b[n])));
        s_att += ac * vm;
        s_xp  += xv * (1.f + att) * vm;
    }
    pool_att[b * DPad + d] = s_att / (float)Nn;
    pool_xp [b * DPad + d] = s_xp;
}

// --------------------------------------------------------------------------
// FC head: D -> DF -> DF -> DF -> 1, relu everywhere (incl. final, as ref)
// one block (128 threads) per batch element
// --------------------------------------------------------------------------
__global__ __launch_bounds__(DFd)
void fc_head(const float* __restrict__ in, const float* __restrict__ W0,
             const float* __restrict__ b0, const float* __restrict__ Wm,
             const float* __restrict__ bm, const float* __restrict__ Wl,
             const float* __restrict__ bl, float* __restrict__ out)
{
    __shared__ float v0[DPad];
    __shared__ float v1[DFd];
    __shared__ float v2[DFd];
    __shared__ float red[DFd];
    const int b = blockIdx.x, t = threadIdx.x;
    for (int f = t; f < Dd; f += DFd) v0[f] = in[b * DPad + f];
    __syncthreads();
    float a = b0[t];
    for (int f = 0; f < Dd; ++f) a += W0[t * Dd + f] * v0[f];
    v1[t] = fmaxf(a, 0.f);
    __syncthreads();
    a = bm[t];
    for (int f = 0; f < DFd; ++f) a += Wm[t * DFd + f] * v1[f];
    v2[t] = fmaxf(a, 0.f);
    __syncthreads();
    a = bm[DFd + t];
    for (int f = 0; f < DFd; ++f) a += Wm[DFd * DFd + t * DFd + f] * v2[f];
    v1[t] = fmaxf(a, 0.f);
    __syncthreads();
    red[t] = Wl[t] * v1[t];
    __syncthreads();
    for (int s = DFd / 2; s > 0; s >>= 1) {
        if (t < s) red[t] += red[t + s];
        __syncthreads();
    }
    if (t == 0) out[b] = fmaxf(red[0] + bl[0], 0.f);
}

// --------------------------------------------------------------------------
// Zero-padded (optionally transposed) weight staging
// dst[drows x dcols]; src[srows x scols]; transpose: dst[r][c]=src[c][r]
// --------------------------------------------------------------------------
__global__ void pad_matrix(const float* __restrict__ src, float* __restrict__ dst,
                           int drows, int dcols, int srows, int scols, int transpose)
{
    int idx = blockIdx.x * blockDim.x + threadIdx.x;
    if (idx >= drows * dcols) return;
    int r = idx / dcols, c = idx % dcols;
    float v = 0.f;
    if (!transpose) { if (r < srows && c < scols) v = src[r * scols + c]; }
    else            { if (r < scols && c < srows) v = src[c * scols + r]; }
    dst[idx] = v;
}

// ==========================================================================
extern "C" void kernel_launch(void* const* d_in, const int* in_sizes, int n_in,
                              void* d_out, int out_size, void* d_ws, size_t ws_size,
                              hipStream_t stream)
{
    (void)in_sizes; (void)n_in; (void)out_size; (void)ws_size;
    const float* c_hs  = (const float*)d_in[0];
    const float* adj1  = (const float*)d_in[1];
    const float* adj2d = (const float*)d_in[2];
    const float* valid = (const float*)d_in[3];
    const float* embW  = (const float*)d_in[4];
    const float* gW    = (const float*)d_in[5];
    const float* gb    = (const float*)d_in[6];
    const float* gA    = (const float*)d_in[7];
    const float* ggw   = (const float*)d_in[8];
    const float* ggb   = (const float*)d_in[9];
    const float* aW    = (const float*)d_in[10];
    const float* ab    = (const float*)d_in[11];
    const float* aA    = (const float*)d_in[12];
    const float* agw   = (const float*)d_in[13];
    const float* agb   = (const float*)d_in[14];
    const float* fcW0  = (const float*)d_in[15];
    const float* fcb0  = (const float*)d_in[16];
    const float* fcWm  = (const float*)d_in[17];
    const float* fcbm  = (const float*)d_in[18];
    const float* fcWl  = (const float*)d_in[19];
    const float* fcbl  = (const float*)d_in[20];
    const float* afW0  = (const float*)d_in[21];
    const float* afb0  = (const float*)d_in[22];
    const float* afWm  = (const float*)d_in[23];
    const float* afbm  = (const float*)d_in[24];
    const float* afWl  = (const float*)d_in[25];
    const float* afbl  = (const float*)d_in[26];
    const float* muP   = (const float*)d_in[27];
    const float* devP  = (const float*)d_in[28];
    const float* bn_g  = (const float*)d_in[29];
    const float* bn_b  = (const float*)d_in[30];
    const float* bn_m  = (const float*)d_in[31];
    const float* bn_v  = (const float*)d_in[32];
    float* out = (float*)d_out;

    // ---- workspace layout (floats) ----
    const long long SZ  = (long long)Bsz * Nn * DPad;   // 2,764,800
    const long long SZE = (long long)Bsz * Nn * Nn;     // 23,040,000
    float* W   = (float*)d_ws;
    float* X   = W;
    float* T1  = X  + SZ;
    float* T2  = T1 + SZ;
    float* Hh  = T2 + SZ;
    float* HA  = Hh + SZ;
    float* HP  = HA + SZ;
    float* E   = HP + SZ;
    float* EMBP = E + SZE;                 // 144*68
    float* GWP  = EMBP + DPad * FINd;      // 4*144*144
    float* GATP = GWP  + NGl * DPad * DPad;
    float* GBP  = GATP + NGl * DPad * DPad; // 4*144
    float* AWP  = GBP  + NGl * DPad;
    float* AATP = AWP  + DPad * DPad;
    float* ABP  = AATP + DPad * DPad;      // 144
    float* PA   = ABP  + DPad;             // pool att: B*DPad
    float* PX   = PA   + Bsz * DPad;

    // ---- stage padded / transposed weights ----
    auto padN = [&](const float* s, float* d, int dr, int dc, int sr, int sc, int tr) {
        int n = dr * dc;
        pad_matrix<<<(n + 255) / 256, 256, 0, stream>>>(s, d, dr, dc, sr, sc, tr);
    };
    padN(embW, EMBP, DPad, FINd, Dd, FINd, 0);
    for (int k = 0; k < NGl; ++k) {
        padN(gW + (long long)k * Dd * Dd, GWP  + (long long)k * DPad * DPad, DPad, DPad, Dd, Dd, 0);
        padN(gA + (long long)k * Dd * Dd, GATP + (long long)k * DPad * DPad, DPad, DPad, Dd, Dd, 1);
        padN(gb + (long long)k * Dd,      GBP  + (long long)k * DPad,        1, DPad, 1, Dd, 0);
    }
    padN(aW, AWP, DPad, DPad, Dd, Dd, 0);
    padN(aA, AATP, DPad, DPad, Dd, Dd, 1);
    padN(ab, ABP, 1, DPad, 1, Dd, 0);

    const dim3 wave(32, 1, 1);
    const dim3 gRows(Bsz * Nn / 16, DPad / 16, 1);   // (1200, 9, 1)
    const dim3 gE(Nn / 16, Nn / 16, Bsz);            // (75, 75, 16)
    const dim3 gHP(Nn / 16, DPad / 16, Bsz);         // (75, 9, 16)

    // ---- embed: X = c_hs @ embW^T  (pad cols -> 0) ----
    wmma_nt_gemm<<<gRows, wave, 0, stream>>>(c_hs, EMBP, X, nullptr,
                                             FINd, FINd, FINd, DPad, 0, 0, 0, 0);

    // one gated-GAT branch:
    //   E = HA*Hh^T + (HA*Hh^T)^T (fused); softmax; HP = relu(E*Hh); gate -> Out
    auto gat_branch = [&](int mode, const float* gwPtr, const float* gbPtr, float* Out) {
        wmma_nt_sym_gemm<<<gE, wave, 0, stream>>>(HA, Hh, E,
                                                  DPad, DPad, Nn,
                                                  (long long)Nn * DPad,
                                                  (long long)Nn * Nn);
        col_softmax<<<dim3((Nn + 31) / 32, Bsz), dim3(32, 8), 0, stream>>>(
            E, adj1, adj2d, muP, devP, mode);
        wmma_nn_gemm<<<gHP, wave, 0, stream>>>(E, Hh, HP,
                                               Nn, Nn, DPad, DPad,
                                               (long long)Nn * Nn, (long long)Nn * DPad,
                                               (long long)Nn * DPad, 1);
        gate_combine<<<Bsz * Nn, wave, 0, stream>>>(X, HP, Out, gwPtr, gbPtr);
    };

    // full layer: Hh = X W^T + b ; HA = Hh A ; both branches ; Diff = T2 - T1
    auto gat_layer = [&](const float* Wp, const float* bp, const float* ATp,
                         const float* gwPtr, const float* gbPtr, float* Diff) {
        wmma_nt_gemm<<<gRows, wave, 0, stream>>>(X, Wp, Hh, bp,
                                                 DPad, DPad, DPad, DPad, 0, 0, 0, 0);
        wmma_nt_gemm<<<gRows, wave, 0, stream>>>(Hh, ATp, HA, nullptr,
                                                 DPad, DPad, DPad, DPad, 0, 0, 0, 0);
        gat_branch(0, gwPtr, gbPtr, T1);   // covalent adjacency
        gat_branch(1, gwPtr, gbPtr, T2);   // RBF-weighted adjacency
        sub_kernel<<<(unsigned)((SZ + 255) / 256), 256, 0, stream>>>(T2, T1, Diff, SZ);
    };

    for (int k = 0; k < NGl; ++k)
        gat_layer(GWP + (long long)k * DPad * DPad, GBP + (long long)k * DPad,
                  GATP + (long long)k * DPad * DPad,
                  ggw + (long long)k * 2 * Dd, ggb + k, X);

    // final attention gate layer -> att_c stored in Hh (after branches done)
    {
        wmma_nt_gemm<<<gRows, wave, 0, stream>>>(X, AWP, Hh, ABP,
                                                 DPad, DPad, DPad, DPad, 0, 0, 0, 0);
        wmma_nt_gemm<<<gRows, wave, 0, stream>>>(Hh, AATP, HA, nullptr,
                                                 DPad, DPad, DPad, DPad, 0, 0, 0, 0);
        gat_branch(0, agw, agb, T1);
        gat_branch(1, agw, agb, T2);
        sub_kernel<<<(unsigned)((SZ + 255) / 256), 256, 0, stream>>>(T2, T1, Hh, SZ);
    }

    // pooling (+BN/sigmoid attention weighting), then the two FC heads
    pool_kernel<<<Bsz, DPad, 0, stream>>>(Hh, X, valid, bn_g, bn_b, bn_m, bn_v, PA, PX);
    fc_head<<<Bsz, DFd, 0, stream>>>(PX, fcW0, fcb0, fcWm, fcbm, fcWl, fcbl, out);
    fc_head<<<Bsz, DFd, 0, stream>>>(PA, afW0, afb0, afWm, afbm, afWl, afbl, out + Bsz);
}